// ReuploadClassifier_19782619366096
// MI455X (gfx1250) — compile-verified
//
#include <hip/hip_runtime.h>
#include <math.h>

#define NQ 7
#define DIM 128
#define NLAYERS 4

typedef __attribute__((ext_vector_type(2))) float v2f;
typedef __attribute__((ext_vector_type(8))) float v8f;

// ---------- complex helpers (float2) ----------
__device__ __forceinline__ float2 cmul(float2 a, float2 b) {
    return make_float2(a.x * b.x - a.y * b.y, a.x * b.y + a.y * b.x);
}
__device__ __forceinline__ float2 cadd(float2 a, float2 b) {
    return make_float2(a.x + b.x, a.y + b.y);
}
// a' = c*own + (-i*s)*partner  (RX mixing; symmetric in the qubit bit)
__device__ __forceinline__ float2 rxmix(float2 o, float2 p, float c, float s) {
    return make_float2(c * o.x + s * p.y, c * o.y - s * p.x);
}
__device__ __forceinline__ float2 shfl_xor_f2(float2 v, int m) {
    float2 r;
    r.x = __shfl_xor(v.x, m, 32);
    r.y = __shfl_xor(v.y, m, 32);
    return r;
}

// =====================================================================
// Kernel 1: quantum circuit simulation. One batch element per wave32.
// State: 128 complex amps; lane owns amps i = 4*lane + slot, slot=0..3.
// Qubit q bit of index i: q=0..4 -> lane bit (4-q); q=5 -> slot bit1;
// q=6 -> slot bit0.
// Writes q_out padded to 8 cols (col 7 = 0) into workspace.
// =====================================================================
__global__ void __launch_bounds__(128)
sim_kernel(const float* __restrict__ x, const float* __restrict__ w,
           float* __restrict__ qout) {
    __shared__ float2 lds[4 * DIM];

    const int lane = threadIdx.x & 31;
    const int wv   = threadIdx.x >> 5;
    const int b    = blockIdx.x * 4 + wv;
    float2* myLds  = lds + wv * DIM;

    // |0...0> initial state
    float2 st[4];
#pragma unroll
    for (int s2 = 0; s2 < 4; ++s2) st[s2] = make_float2(0.f, 0.f);
    if (lane == 0) st[0].x = 1.0f;

    // RX angles are re-uploaded identically every layer: trig once.
    float cx[NQ], sx[NQ];
#pragma unroll
    for (int q = 0; q < NQ; ++q) {
        const float xv = x[b * NQ + q];
        __sincosf(0.5f * xv, &sx[q], &cx[q]);
    }

    // Composite CNOT-ring permutation (same every layer): new[i] = old[jp[i]]
    int jp[4];
#pragma unroll
    for (int s2 = 0; s2 < 4; ++s2) {
        int j = 4 * lane + s2;
#pragma unroll
        for (int q = NQ - 1; q >= 0; --q) {
            const int c = q, t = (q + 1) % NQ;
            const int ctrl = (j >> (NQ - 1 - c)) & 1;
            j = ctrl ? (j ^ (1 << (NQ - 1 - t))) : j;
        }
        jp[s2] = j;
    }

    for (int l = 0; l < NLAYERS; ++l) {
        // ---- RX(x_q) data-reupload gates ----
#pragma unroll
        for (int q = 0; q < NQ; ++q) {
            const float c = cx[q], s = sx[q];
            if (q < 5) {
                const int m = 16 >> q;
                float2 p[4];
#pragma unroll
                for (int s2 = 0; s2 < 4; ++s2) p[s2] = shfl_xor_f2(st[s2], m);
#pragma unroll
                for (int s2 = 0; s2 < 4; ++s2) st[s2] = rxmix(st[s2], p[s2], c, s);
            } else if (q == 5) {  // partner slot ^ 2
                float2 n0 = rxmix(st[0], st[2], c, s);
                float2 n1 = rxmix(st[1], st[3], c, s);
                float2 n2 = rxmix(st[2], st[0], c, s);
                float2 n3 = rxmix(st[3], st[1], c, s);
                st[0] = n0; st[1] = n1; st[2] = n2; st[3] = n3;
            } else {              // q == 6, partner slot ^ 1
                float2 n0 = rxmix(st[0], st[1], c, s);
                float2 n1 = rxmix(st[1], st[0], c, s);
                float2 n2 = rxmix(st[2], st[3], c, s);
                float2 n3 = rxmix(st[3], st[2], c, s);
                st[0] = n0; st[1] = n1; st[2] = n2; st[3] = n3;
            }
        }

        // ---- Rot(phi,theta,omega) trainable gates ----
#pragma unroll
        for (int q = 0; q < NQ; ++q) {
            const float phi = w[(l * NQ + q) * 3 + 0];
            const float th  = w[(l * NQ + q) * 3 + 1];
            const float om  = w[(l * NQ + q) * 3 + 2];
            float stq, ctq; __sincosf(0.5f * th, &stq, &ctq);
            float sa, ca;   __sincosf(0.5f * (phi + om), &sa, &ca);
            float sb, cb;   __sincosf(0.5f * (phi - om), &sb, &cb);
            const float2 u00 = make_float2( ctq * ca, -ctq * sa);
            const float2 u01 = make_float2(-stq * cb, -stq * sb);
            const float2 u10 = make_float2( stq * cb, -stq * sb);
            const float2 u11 = make_float2( ctq * ca,  ctq * sa);
            if (q < 5) {
                const int m  = 16 >> q;
                const int bq = (lane >> (4 - q)) & 1;
                const float2 cO = bq ? u11 : u00;
                const float2 cP = bq ? u10 : u01;
                float2 p[4];
#pragma unroll
                for (int s2 = 0; s2 < 4; ++s2) p[s2] = shfl_xor_f2(st[s2], m);
#pragma unroll
                for (int s2 = 0; s2 < 4; ++s2)
                    st[s2] = cadd(cmul(cO, st[s2]), cmul(cP, p[s2]));
            } else if (q == 5) {
                float2 n0 = cadd(cmul(u00, st[0]), cmul(u01, st[2]));
                float2 n1 = cadd(cmul(u00, st[1]), cmul(u01, st[3]));
                float2 n2 = cadd(cmul(u10, st[0]), cmul(u11, st[2]));
                float2 n3 = cadd(cmul(u10, st[1]), cmul(u11, st[3]));
                st[0] = n0; st[1] = n1; st[2] = n2; st[3] = n3;
            } else {
                float2 n0 = cadd(cmul(u00, st[0]), cmul(u01, st[1]));
                float2 n1 = cadd(cmul(u10, st[0]), cmul(u11, st[1]));
                float2 n2 = cadd(cmul(u00, st[2]), cmul(u01, st[3]));
                float2 n3 = cadd(cmul(u10, st[2]), cmul(u11, st[3]));
                st[0] = n0; st[1] = n1; st[2] = n2; st[3] = n3;
            }
        }

        // ---- CNOT ring: permuted gather via per-wave LDS slab ----
        __syncthreads();
#pragma unroll
        for (int s2 = 0; s2 < 4; ++s2) myLds[4 * lane + s2] = st[s2];
        __syncthreads();
#pragma unroll
        for (int s2 = 0; s2 < 4; ++s2) st[s2] = myLds[jp[s2]];
    }

    // ---- <Z_q> readout: probs dot Z_SIGNS ----
    float acc[NQ];
#pragma unroll
    for (int q = 0; q < NQ; ++q) acc[q] = 0.f;
#pragma unroll
    for (int s2 = 0; s2 < 4; ++s2) {
        const int i = 4 * lane + s2;
        const float p = st[s2].x * st[s2].x + st[s2].y * st[s2].y;
#pragma unroll
        for (int q = 0; q < NQ; ++q)
            acc[q] += ((i >> (NQ - 1 - q)) & 1) ? -p : p;
    }
#pragma unroll
    for (int m = 16; m >= 1; m >>= 1) {
#pragma unroll
        for (int q = 0; q < NQ; ++q) acc[q] += __shfl_xor(acc[q], m, 32);
    }
    if (lane < NQ) qout[b * 8 + lane] = acc[lane];
    if (lane == NQ) qout[b * 8 + 7] = 0.f;  // K padding for WMMA
}

// =====================================================================
// Kernel 2: MLP head with V_WMMA_F32_16X16X4_F32. One wave per 16-row
// batch tile. GEMM1 (K=7 pad 8) -> bias+ReLU -> LDS restage -> GEMM2
// (K=32, N=22 pad 32) -> bias -> store.
// Fragment layouts (ISA 7.12.2): A lanes0-15 K={k0,k0+1}, lanes16-31
// K={k0+2,k0+3}; B VGPR0/1 hold K rows {k0,k0+1} (lo half) / {k0+2,k0+3}
// (hi half); C/D: VGPR r -> M = r + 8*(lane>=16), N = lane&15.
// =====================================================================
__global__ void __launch_bounds__(128)
mlp_kernel(const float* __restrict__ Q,        // B x 8 (padded q_out)
           const float* __restrict__ w1,       // 32 x 7
           const float* __restrict__ b1,       // 32
           const float* __restrict__ w2,       // 22 x 32
           const float* __restrict__ b2,       // 22
           float* __restrict__ out) {          // B x 22
    __shared__ float hbuf[4][16 * 32];

    const int lane = threadIdx.x & 31;
    const int wv   = threadIdx.x >> 5;
    const int tile = blockIdx.x * 4 + wv;
    const int half = lane >> 4;
    const int rl   = lane & 15;
    const int row  = tile * 16 + rl;   // batch row for A fragments

    // ---------- GEMM1: h = relu(Q @ W1^T + b1), 16 x 32 ----------
#pragma unroll
    for (int n = 0; n < 2; ++n) {
        const int col = n * 16 + rl;   // hidden unit index (< 32)
        v8f c = {0.f, 0.f, 0.f, 0.f, 0.f, 0.f, 0.f, 0.f};
#pragma unroll
        for (int kk = 0; kk < 2; ++kk) {
            const int k0 = kk * 4 + half * 2;
            v2f a;
            a.x = Q[row * 8 + k0];
            a.y = Q[row * 8 + k0 + 1];
            v2f bb;
            bb.x = (k0     < NQ) ? w1[col * NQ + k0]     : 0.f;
            bb.y = (k0 + 1 < NQ) ? w1[col * NQ + k0 + 1] : 0.f;
            c = __builtin_amdgcn_wmma_f32_16x16x4_f32(false, a, false, bb,
                                                      (short)0, c, false, false);
        }
        const float bias = b1[col];
#pragma unroll
        for (int r = 0; r < 8; ++r) {
            const float h = fmaxf(c[r] + bias, 0.f);
            hbuf[wv][(r + 8 * half) * 32 + col] = h;
        }
    }
    __syncthreads();

    // ---------- GEMM2: out = h @ W2^T + b2, 16 x 22 ----------
#pragma unroll
    for (int n = 0; n < 2; ++n) {
        const int col = n * 16 + rl;   // output class index (valid if < 22)
        v8f c = {0.f, 0.f, 0.f, 0.f, 0.f, 0.f, 0.f, 0.f};
#pragma unroll
        for (int kk = 0; kk < 8; ++kk) {
            const int k0 = kk * 4 + half * 2;
            v2f a;
            a.x = hbuf[wv][rl * 32 + k0];
            a.y = hbuf[wv][rl * 32 + k0 + 1];
            v2f bb;
            bb.x = (col < 22) ? w2[col * 32 + k0]     : 0.f;
            bb.y = (col < 22) ? w2[col * 32 + k0 + 1] : 0.f;
            c = __builtin_amdgcn_wmma_f32_16x16x4_f32(false, a, false, bb,
                                                      (short)0, c, false, false);
        }
        if (col < 22) {
            const float bias = b2[col];
#pragma unroll
            for (int r = 0; r < 8; ++r) {
                const int m = tile * 16 + r + 8 * half;
                out[m * 22 + col] = c[r] + bias;
            }
        }
    }
}

extern "C" void kernel_launch(void* const* d_in, const int* in_sizes, int n_in,
                              void* d_out, int out_size, void* d_ws, size_t ws_size,
                              hipStream_t stream) {
    const float* x    = (const float*)d_in[0];  // (B, 7)
    const float* w    = (const float*)d_in[1];  // (4, 7, 3)
    const float* fc1w = (const float*)d_in[2];  // (32, 7)
    const float* fc1b = (const float*)d_in[3];  // (32,)
    const float* fc2w = (const float*)d_in[4];  // (22, 32)
    const float* fc2b = (const float*)d_in[5];  // (22,)
    float* out  = (float*)d_out;                // (B, 22)
    float* qout = (float*)d_ws;                 // (B, 8) padded q_out

    const int B = in_sizes[0] / NQ;             // 16384

    // One batch element per wave, 4 waves per block.
    sim_kernel<<<B / 4, 128, 0, stream>>>(x, w, qout);
    // One 16-row tile per wave, 4 waves per block.
    mlp_kernel<<<(B / 16) / 4, 128, 0, stream>>>(qout, fc1w, fc1b, fc2w, fc2b, out);
}